// HilbertSerialization_33689723470048
// MI455X (gfx1250) — compile-verified
//
#include <hip/hip_runtime.h>
#include <hip/hip_bf16.h>

#define RADIX        256
#define SCAT_THREADS 256
#define SUBTILES     8
#define TILE         (SCAT_THREADS * SUBTILES)   // 2048 keys per block per pass

// ---------------------------------------------------------------------------
// gfx1250 async global->LDS helpers (ASYNCcnt path, see CDNA5 ISA)
// ---------------------------------------------------------------------------
__device__ __forceinline__ void async_copy_b64_to_lds(void* lds_dst, const void* gsrc) {
  // LDS generic address: low 32 bits are the LDS byte offset (ISA 10.2 aperture rules).
  unsigned lds_off = (unsigned)(size_t)lds_dst;
  asm volatile("global_load_async_to_lds_b64 %0, %1, off"
               :: "v"(lds_off), "v"(gsrc)
               : "memory");
}

__device__ __forceinline__ void wait_async0() {
#if __has_builtin(__builtin_amdgcn_s_wait_asynccnt)
  __builtin_amdgcn_s_wait_asynccnt(0);
#else
  asm volatile("s_wait_asynccnt 0x0" ::: "memory");
#endif
}

// ---------------------------------------------------------------------------
// K0: Hilbert key build.  key = (hilbert26 << 21) | original_index
// ---------------------------------------------------------------------------
__global__ void hs_build_keys(const int* __restrict__ coords,
                              const long long* __restrict__ sshape,
                              const int* __restrict__ shifts,
                              unsigned long long* __restrict__ keys, int n) {
  int i = blockIdx.x * blockDim.x + threadIdx.x;
  if (i >= n) return;
  const int D = (int)sshape[0], H = (int)sshape[1], W = (int)sshape[2];
  const int sh = shifts[0];
  const int sx = sh ? 15 : 0;   // ceil(30/2)
  const int sy = sh ? 15 : 0;   // ceil(30/2)
  const int sz = sh ? 4  : 0;   // ceil(8/2)

  const int b = coords[4 * i + 0];
  const int z = coords[4 * i + 1];
  const int y = coords[4 * i + 2];
  const int x = coords[4 * i + 3];

  unsigned X[3];
  X[0] = ((unsigned)((x + sx) % W)) & 255u;  // reference drops bits above DEPTH=8
  X[1] = ((unsigned)((y + sy) % H)) & 255u;
  X[2] = ((unsigned)((z + sz) % D)) & 255u;

  // Skilling transform (matches the reference's per-bit/per-dim loop exactly):
  // bit q set  -> invert lower bits of X[0]
  // bit q clear-> swap lower bits of X[0] and X[dim]
#pragma unroll
  for (int q = 7; q >= 1; --q) {
    const unsigned low = (1u << q) - 1u;
#pragma unroll
    for (int dim = 0; dim < 3; ++dim) {
      if ((X[dim] >> q) & 1u) {
        X[0] ^= low;
      } else {
        unsigned t = (X[0] ^ X[dim]) & low;
        X[0] ^= t;
        X[dim] ^= t;
      }
    }
  }

  // Interleave MSB-first: position = bit*3 + dim (dim0 = x most significant in triple)
  unsigned g = 0;
#pragma unroll
  for (int bit = 0; bit < 8; ++bit) {
    const int q = 7 - bit;
    g = (g << 3) | (((X[0] >> q) & 1u) << 2) | (((X[1] >> q) & 1u) << 1) | ((X[2] >> q) & 1u);
  }

  // Gray -> binary over the 24-bit sequence (shift chain 16,8,4,2,1 as in reference)
  unsigned h = g;
  h ^= h >> 16; h ^= h >> 8; h ^= h >> 4; h ^= h >> 2; h ^= h >> 1;

  const unsigned code = h | ((unsigned)b << 24);     // 26 significant bits
  keys[i] = ((unsigned long long)code << 21) | (unsigned)i;
}

// ---------------------------------------------------------------------------
// K1: per-block digit histogram (digit-major global layout: hist[d*numBlocks+blk])
// ---------------------------------------------------------------------------
__global__ void hs_radix_hist(const unsigned long long* __restrict__ keys, int n,
                              unsigned int* __restrict__ hist, int numBlocks, int shift) {
  __shared__ unsigned int h[RADIX];
  for (int d = threadIdx.x; d < RADIX; d += blockDim.x) h[d] = 0u;
  __syncthreads();
  const int base = blockIdx.x * TILE;
  for (int t = threadIdx.x; t < TILE; t += blockDim.x) {
    const int i = base + t;
    if (i < n) {
      const unsigned d = (unsigned)(keys[i] >> shift) & 255u;
      atomicAdd(&h[d], 1u);
    }
  }
  __syncthreads();
  for (int d = threadIdx.x; d < RADIX; d += blockDim.x)
    hist[(size_t)d * numBlocks + blockIdx.x] = h[d];
}

// ---------------------------------------------------------------------------
// K2a: per-digit totals (one block per digit)
// ---------------------------------------------------------------------------
__global__ void hs_digit_totals(const unsigned int* __restrict__ hist, int numBlocks,
                                unsigned int* __restrict__ totals) {
  __shared__ unsigned int red[256];
  const unsigned int* p = hist + (size_t)blockIdx.x * numBlocks;
  unsigned int s = 0;
  for (int i = threadIdx.x; i < numBlocks; i += blockDim.x) s += p[i];
  red[threadIdx.x] = s;
  __syncthreads();
  for (int off = 128; off > 0; off >>= 1) {
    if ((int)threadIdx.x < off) red[threadIdx.x] += red[threadIdx.x + off];
    __syncthreads();
  }
  if (threadIdx.x == 0) totals[blockIdx.x] = red[0];
}

// ---------------------------------------------------------------------------
// K2b: exclusive scan of the 256 digit totals (single block)
// ---------------------------------------------------------------------------
__global__ void hs_scan_totals(const unsigned int* __restrict__ totals,
                               unsigned int* __restrict__ bases) {
  __shared__ unsigned int tmp[256];
  const unsigned int orig = totals[threadIdx.x];
  tmp[threadIdx.x] = orig;
  __syncthreads();
  for (int off = 1; off < 256; off <<= 1) {
    unsigned int v = ((int)threadIdx.x >= off) ? tmp[threadIdx.x - off] : 0u;
    __syncthreads();
    tmp[threadIdx.x] += v;
    __syncthreads();
  }
  bases[threadIdx.x] = tmp[threadIdx.x] - orig;   // exclusive
}

// ---------------------------------------------------------------------------
// K2c: per-digit exclusive scan over blocks, + digit base (one block per digit)
// ---------------------------------------------------------------------------
__global__ void hs_scan_block_offsets(unsigned int* __restrict__ hist, int numBlocks,
                                      const unsigned int* __restrict__ bases) {
  __shared__ unsigned int tmp[256];
  unsigned int* p = hist + (size_t)blockIdx.x * numBlocks;
  unsigned int running = bases[blockIdx.x];
  for (int c = 0; c < numBlocks; c += 256) {
    const int i = c + (int)threadIdx.x;
    const unsigned int v = (i < numBlocks) ? p[i] : 0u;
    tmp[threadIdx.x] = v;
    __syncthreads();
    for (int off = 1; off < 256; off <<= 1) {
      unsigned int u = ((int)threadIdx.x >= off) ? tmp[threadIdx.x - off] : 0u;
      __syncthreads();
      tmp[threadIdx.x] += u;
      __syncthreads();
    }
    const unsigned int incl = tmp[threadIdx.x];
    const unsigned int chunkTotal = tmp[255];
    __syncthreads();                      // before tmp is overwritten next chunk
    if (i < numBlocks) p[i] = incl - v + running;
    running += chunkTotal;
  }
}

// ---------------------------------------------------------------------------
// K3: stable scatter.  256 thr = 8 wave32; 8 sub-tiles of 256 keys, async-staged
// through LDS (double buffered).  Stability: lane order -> wave order -> sub-tile
// order -> block order (digit-major global scan).
// ---------------------------------------------------------------------------
__global__ void hs_radix_scatter(const unsigned long long* __restrict__ src,
                                 unsigned long long* __restrict__ dst, int n,
                                 const unsigned int* __restrict__ blockOffs,
                                 int numBlocks, int shift) {
  __shared__ unsigned int ldsBase[RADIX];            // running scatter base per digit
  __shared__ unsigned int waveHist[8][RADIX];        // per-wave digit counts / prefixes
  __shared__ unsigned int subTotal[RADIX];           // sub-tile digit totals
  __shared__ unsigned long long stage[2][SCAT_THREADS];

  const int tid  = (int)threadIdx.x;
  const int lane = tid & 31;
  const int wave = tid >> 5;
  const int blk  = (int)blockIdx.x;
  const int base = blk * TILE;

  for (int d = tid; d < RADIX; d += SCAT_THREADS)
    ldsBase[d] = blockOffs[(size_t)d * numBlocks + blk];

  // kick off async staging of sub-tile 0
  {
    const int i = base + tid;
    if (i < n) async_copy_b64_to_lds(&stage[0][tid], &src[i]);
  }

  int buf = 0;
  for (int s = 0; s < SUBTILES; ++s) {
    wait_async0();                                   // this wave's staged keys are in LDS
    const int i = base + s * SCAT_THREADS + tid;
    const bool valid = (i < n);
    const unsigned long long k = stage[buf][tid];    // ds_load_b64

    // prefetch next sub-tile into the other buffer (overlaps with ranking below)
    if (s + 1 < SUBTILES) {
      const int ni = base + (s + 1) * SCAT_THREADS + tid;
      if (ni < n) async_copy_b64_to_lds(&stage[buf ^ 1][tid], &src[ni]);
    }

    const unsigned digit = (unsigned)(k >> shift) & 255u;

    __syncthreads();   // previous sub-tile fully done with waveHist/subTotal/ldsBase

    // zero per-wave histograms (8*256 entries / 256 threads)
    unsigned int* wh = &waveHist[0][0];
    for (int j = tid; j < 8 * RADIX; j += SCAT_THREADS) wh[j] = 0u;
    __syncthreads();

    // wave32 peer matching: lanes with identical digit
    unsigned peers = __builtin_amdgcn_ballot_w32(valid);
#pragma unroll
    for (int bp = 0; bp < 8; ++bp) {
      const unsigned bset = __builtin_amdgcn_ballot_w32(valid && ((digit >> bp) & 1u));
      peers &= ((digit >> bp) & 1u) ? bset : ~bset;
    }
    const unsigned lt = (lane == 0) ? 0u : ((1u << lane) - 1u);
    const unsigned rankInWave = (unsigned)__popc(peers & lt);
    if (valid && (peers & lt) == 0u)                 // lowest valid lane of peer group
      waveHist[wave][digit] = (unsigned)__popc(peers);
    __syncthreads();

    // per-digit exclusive prefix across the 8 waves (one digit per thread)
    {
      unsigned run = 0;
#pragma unroll
      for (int w = 0; w < 8; ++w) {
        const unsigned t = waveHist[w][tid];
        waveHist[w][tid] = run;
        run += t;
      }
      subTotal[tid] = run;
    }
    __syncthreads();

    if (valid) {
      const unsigned pos = ldsBase[digit] + waveHist[wave][digit] + rankInWave;
      dst[pos] = k;
    }
    __syncthreads();                                 // all ldsBase reads done
    ldsBase[tid] += subTotal[tid];                   // advance for next sub-tile

    buf ^= 1;
  }
}

// ---------------------------------------------------------------------------
// K4: emit argsort result (indices exact in f32 since N < 2^24)
// ---------------------------------------------------------------------------
__global__ void hs_write_out(const unsigned long long* __restrict__ keys,
                             float* __restrict__ out, int n) {
  const int i = blockIdx.x * blockDim.x + threadIdx.x;
  if (i < n) out[i] = (float)(unsigned)(keys[i] & 0x1FFFFFull);
}

// ---------------------------------------------------------------------------
extern "C" void kernel_launch(void* const* d_in, const int* in_sizes, int n_in,
                              void* d_out, int out_size, void* d_ws, size_t ws_size,
                              hipStream_t stream) {
  const int*       coords = (const int*)d_in[0];        // [N,4] int32 (b,z,y,x)
  const long long* sshape = (const long long*)d_in[1];  // [3] int64 (D,H,W)
  const int*       shifts = (const int*)d_in[2];        // scalar int

  const int n = in_sizes[0] / 4;
  const int numBlocks = (n + TILE - 1) / TILE;

  char* ws = (char*)d_ws;
  unsigned long long* keysA  = (unsigned long long*)ws;
  unsigned long long* keysB  = keysA + n;
  unsigned int*       hist   = (unsigned int*)(keysB + n);
  unsigned int*       totals = hist + (size_t)RADIX * numBlocks;
  unsigned int*       bases  = totals + RADIX;

  const dim3 b256(256);
  hs_build_keys<<<(n + 255) / 256, b256, 0, stream>>>(coords, sshape, shifts, keysA, n);

  unsigned long long* src = keysA;
  unsigned long long* dst = keysB;
  for (int p = 0; p < 4; ++p) {                       // code occupies key bits [21,47)
    const int shift = 21 + 8 * p;
    hs_radix_hist<<<numBlocks, b256, 0, stream>>>(src, n, hist, numBlocks, shift);
    hs_digit_totals<<<RADIX, b256, 0, stream>>>(hist, numBlocks, totals);
    hs_scan_totals<<<1, b256, 0, stream>>>(totals, bases);
    hs_scan_block_offsets<<<RADIX, b256, 0, stream>>>(hist, numBlocks, bases);
    hs_radix_scatter<<<numBlocks, b256, 0, stream>>>(src, dst, n, hist, numBlocks, shift);
    unsigned long long* t = src; src = dst; dst = t;
  }
  // 4 passes -> final keys back in keysA (== src after last swap)
  hs_write_out<<<(n + 255) / 256, b256, 0, stream>>>(src, (float*)d_out, n);
}